// GCNNodeRegressor_59708635349185
// MI455X (gfx1250) — compile-verified
//
#include <hip/hip_runtime.h>

typedef __attribute__((ext_vector_type(2))) float v2f;
typedef __attribute__((ext_vector_type(8))) float v8f;

#define N_NODES 100000
#define N_EDGES 3200000
#define IN_CONT 11
#define EMB_DIM 8
#define KPAD    20      // 19 features padded to multiple of 4
#define HIDDEN  64
#define MTILES  (N_NODES / 16)   // 6250, N divisible by 16

// ---------------- degree ----------------
__global__ void k_init_deg(float* deg) {
    int i = blockIdx.x * blockDim.x + threadIdx.x;
    if (i < N_NODES) deg[i] = 1.0f;               // self loop
}

__global__ void k_accum_deg(const int* __restrict__ dst, float* deg) {
    int e = blockIdx.x * blockDim.x + threadIdx.x;
    if (e < N_EDGES) atomicAdd(&deg[dst[e]], 1.0f);
}

__global__ void k_rsqrt_deg(float* deg) {
    int i = blockIdx.x * blockDim.x + threadIdx.x;
    if (i < N_NODES) deg[i] = rsqrtf(deg[i]);     // deg >= 1 always (self loop)
}

// ---------------- feature build: [N, 20] = [cont(11) | emb(8) | 0] ----------------
__global__ void k_build_feat(const float* __restrict__ x_cont,
                             const int* __restrict__ func_type,
                             const float* __restrict__ emb_table,
                             float* __restrict__ feat) {
    int i = blockIdx.x * blockDim.x + threadIdx.x;
    if (i >= N_NODES) return;
    const float* xc = x_cont + (long)i * IN_CONT;
    const float* em = emb_table + func_type[i] * EMB_DIM;
    float* o = feat + (long)i * KPAD;
    #pragma unroll
    for (int j = 0; j < IN_CONT; ++j) o[j] = xc[j];
    #pragma unroll
    for (int j = 0; j < EMB_DIM; ++j) o[IN_CONT + j] = em[j];
    o[KPAD - 1] = 0.0f;
}

// ---------------- pad W1 [19,64] -> Wp [20,64] with zero last row ----------------
__global__ void k_pad_w1(const float* __restrict__ W1, float* __restrict__ Wp) {
    int i = blockIdx.x * blockDim.x + threadIdx.x;
    if (i >= KPAD * HIDDEN) return;
    Wp[i] = ((i >> 6) < 19) ? W1[i] : 0.0f;
}

// ---------------- generic zero ----------------
__global__ void k_zero(float* p, int n) {
    int i = blockIdx.x * blockDim.x + threadIdx.x;
    if (i < n) p[i] = 0.0f;
}

// ---------------- GEMM: C[M,64] = A[M,LDA] @ B[LDA,64], fp32 WMMA 16x16x4 ----------------
// One wave computes a 16x64 output tile with 4 v8f accumulators. Fully unrolled K loop.
// A frag (16x4 fp32): lane L<16 holds row L, K = k0+{0,1}; lane L>=16 holds row L-16, K = k0+{2,3}.
// B frag (4x16 fp32): lane L<16 holds col L, K = k0+{0,1}; lane L>=16 holds col L-16, K = k0+{2,3}.
// C/D (16x16 fp32): VGPR v -> row v (lanes 0-15) / row v+8 (lanes 16-31), lane&15 -> col.
template <int LDA, int KSTEPS>
__global__ void k_gemm_wmma(const float* __restrict__ A,
                            const float* __restrict__ B,
                            float* __restrict__ C, int ntiles) {
    int wave = (blockIdx.x * blockDim.x + threadIdx.x) >> 5;
    int lane = threadIdx.x & 31;
    if (wave >= ntiles) return;                    // wave-uniform: EXEC stays all-1s
    int rc    = lane & 15;                         // row (A) / col (B)
    int khalf = (lane >> 4) * 2;                   // 0 or 2

    const float* Arow = A + (long)(wave * 16 + rc) * LDA + khalf;
    const float* Bcol = B + (long)khalf * HIDDEN + rc;
    v8f acc[4] = {v8f{}, v8f{}, v8f{}, v8f{}};

    #pragma unroll
    for (int s = 0; s < KSTEPS; ++s) {
        v2f a;
        a.x = Arow[s * 4 + 0];
        a.y = Arow[s * 4 + 1];
        #pragma unroll
        for (int nt = 0; nt < 4; ++nt) {
            v2f b;
            b.x = Bcol[(s * 4 + 0) * HIDDEN + nt * 16];
            b.y = Bcol[(s * 4 + 1) * HIDDEN + nt * 16];
            acc[nt] = __builtin_amdgcn_wmma_f32_16x16x4_f32(
                false, a, false, b, (short)0, acc[nt], false, false);
        }
    }

    int mrow = (lane >> 4) * 8;
    #pragma unroll
    for (int nt = 0; nt < 4; ++nt)
        #pragma unroll
        for (int v = 0; v < 8; ++v)
            C[(long)(wave * 16 + mrow + v) * HIDDEN + nt * 16 + rc] = acc[nt][v];
}

// ---------------- edge scatter: out[dst] += xw[src] * (dis[src]*dis[dst]) ----------------
// 16 threads per edge, float4 each (64 floats total). Index loads broadcast per wave.
__global__ void k_edge_scatter(const float* __restrict__ xw,
                               const int* __restrict__ src,
                               const int* __restrict__ dst,
                               const float* __restrict__ dis,
                               float* __restrict__ out) {
    int tid = blockIdx.x * blockDim.x + threadIdx.x;   // E*16 = 51.2M < 2^31
    int e = tid >> 4;
    int c = tid & 15;
    if (e >= N_EDGES) return;
    int s = src[e], d = dst[e];
    float norm = dis[s] * dis[d];
    const float4* xv = (const float4*)(xw + (long)s * HIDDEN);
    float4 v = xv[c];
    float* o = out + (long)d * HIDDEN + c * 4;
    atomicAdd(o + 0, v.x * norm);
    atomicAdd(o + 1, v.y * norm);
    atomicAdd(o + 2, v.z * norm);
    atomicAdd(o + 3, v.w * norm);
}

// ---------------- h = relu(agg + xw*dis^2 + b), in place on agg ----------------
__global__ void k_bias_relu_self(float* __restrict__ agg,
                                 const float* __restrict__ xw,
                                 const float* __restrict__ dis,
                                 const float* __restrict__ b) {
    int idx = blockIdx.x * blockDim.x + threadIdx.x;
    if (idx >= N_NODES * HIDDEN) return;
    int i = idx >> 6;
    int f = idx & 63;
    float d = dis[i];
    float h = agg[idx] + xw[idx] * d * d + b[f];
    agg[idx] = fmaxf(h, 0.0f);
}

// ---------------- final: out[i] = relu(agg + xw*dis^2 + b2) . lin_W + lin_b ----------------
// One wave per node; 2 features per lane; wave32 shuffle reduction.
__global__ void k_final(const float* __restrict__ agg,
                        const float* __restrict__ xw,
                        const float* __restrict__ dis,
                        const float* __restrict__ b2,
                        const float* __restrict__ linW,
                        const float* __restrict__ linb,
                        float* __restrict__ out) {
    int wave = (blockIdx.x * blockDim.x + threadIdx.x) >> 5;
    int lane = threadIdx.x & 31;
    if (wave >= N_NODES) return;
    float d = dis[wave];
    float d2 = d * d;
    const float* ag = agg + (long)wave * HIDDEN;
    const float* xv = xw + (long)wave * HIDDEN;
    float sum = 0.0f;
    #pragma unroll
    for (int t = 0; t < 2; ++t) {
        int f = lane + t * 32;
        float h = fmaxf(ag[f] + xv[f] * d2 + b2[f], 0.0f);
        sum += h * linW[f];
    }
    #pragma unroll
    for (int off = 16; off > 0; off >>= 1)
        sum += __shfl_xor(sum, off, 32);
    if (lane == 0) out[wave] = sum + linb[0];
}

extern "C" void kernel_launch(void* const* d_in, const int* in_sizes, int n_in,
                              void* d_out, int out_size, void* d_ws, size_t ws_size,
                              hipStream_t stream) {
    const float* x_cont    = (const float*)d_in[0];
    const int*   func_type = (const int*)  d_in[1];
    const int*   edge      = (const int*)  d_in[2];   // [2,E]: row0=src, row1=dst
    const float* emb_table = (const float*)d_in[3];
    const float* W1        = (const float*)d_in[4];
    const float* b1        = (const float*)d_in[5];
    const float* W2        = (const float*)d_in[6];
    const float* b2        = (const float*)d_in[7];
    const float* linW      = (const float*)d_in[8];
    const float* linb      = (const float*)d_in[9];
    float* out = (float*)d_out;

    const int* src = edge;
    const int* dst = edge + N_EDGES;

    // workspace layout (floats); all float4-aligned
    float* ws    = (float*)d_ws;
    float* deg   = ws;                                  // N
    float* feat  = ws + 100000;                         // N*20
    float* bufA  = ws + 2100000;                        // N*64 (xw1 / xw2)
    float* bufB  = ws + 8500000;                        // N*64 (agg1 -> h1)
    float* bufC  = ws + 14900000;                       // N*64 (agg2)
    float* W1p   = ws + 21300000;                       // 20*64 padded W1

    const int NB_NODE = (N_NODES + 255) / 256;          // 391
    const int NB_EDGE = (N_EDGES + 255) / 256;          // 12500
    const int NB_H    = (N_NODES * HIDDEN) / 256;       // 25000
    const int NB_SC   = (N_EDGES * 16) / 256;           // 200000
    const int NB_GEMM = (MTILES + 7) / 8;               // 782 (8 waves/block)
    const int NB_FIN  = (N_NODES + 7) / 8;              // 12500

    // degrees -> deg_inv_sqrt
    k_init_deg <<<NB_NODE, 256, 0, stream>>>(deg);
    k_accum_deg<<<NB_EDGE, 256, 0, stream>>>(dst, deg);
    k_rsqrt_deg<<<NB_NODE, 256, 0, stream>>>(deg);

    // features + padded W1
    k_build_feat<<<NB_NODE, 256, 0, stream>>>(x_cont, func_type, emb_table, feat);
    k_pad_w1<<<(KPAD * HIDDEN + 255) / 256, 256, 0, stream>>>(W1, W1p);

    // layer 1: xw1 = feat @ W1p   (K = 20 padded)
    k_gemm_wmma<KPAD, KPAD / 4><<<NB_GEMM, 256, 0, stream>>>(feat, W1p, bufA, MTILES);
    k_zero<<<NB_H, 256, 0, stream>>>(bufB, N_NODES * HIDDEN);
    k_edge_scatter<<<NB_SC, 256, 0, stream>>>(bufA, src, dst, deg, bufB);
    k_bias_relu_self<<<NB_H, 256, 0, stream>>>(bufB, bufA, deg, b1);   // bufB = h1

    // layer 2: xw2 = h1 @ W2  (K = 64)
    k_gemm_wmma<HIDDEN, HIDDEN / 4><<<NB_GEMM, 256, 0, stream>>>(bufB, W2, bufA, MTILES);
    k_zero<<<NB_H, 256, 0, stream>>>(bufC, N_NODES * HIDDEN);
    k_edge_scatter<<<NB_SC, 256, 0, stream>>>(bufA, src, dst, deg, bufC);

    // fused bias+relu+linear head
    k_final<<<NB_FIN, 256, 0, stream>>>(bufC, bufA, deg, b2, linW, linb, out);
}